// MultimodalMambaModel_33801392619926
// MI455X (gfx1250) — compile-verified
//
#include <hip/hip_runtime.h>
#include <hip/hip_bf16.h>
#include <math.h>

// ---------------------------------------------------------------------------
// MultimodalMamba forward on gfx1250 (CDNA5, wave32, WMMA).
// All dense GEMMs use V_WMMA_F32_16X16X4_F32 (full fp32 matrix path).
// ---------------------------------------------------------------------------

#define D_MODEL   384
#define D_INNER   768
#define D_STATE   16
#define DT_RANK   24
#define D_CONV    4
#define VOCABSZ   50257
#define AUDIO_DIM 80
#define BATCH     2
#define L_AUDIO   1024
#define L_TEXT    1024
#define L_TOT     2048
#define ROWS      (BATCH * L_TOT)      // 4096

typedef float v2f __attribute__((ext_vector_type(2)));
typedef float v8f __attribute__((ext_vector_type(8)));

// ---------------------------------------------------------------------------
// Generic fp32 WMMA GEMM:  C[M,N] = act( A[M,:K (stride lda)] @ W[N,K]^T + bias )
// One wave -> 32x64 macro-tile (2 M-subtiles x 4 N-subtiles -> 8 WMMA per
// K-step of 4 against 6 b64 loads). K % 4 == 0, M % 32 == 0. N tail handled
// by clamped loads + masked stores (wave-uniform branches only -> EXEC
// all-ones at every WMMA, as the ISA requires).
// act: 0 = none, 1 = softplus (for dt_proj)
// ---------------------------------------------------------------------------
__global__ __launch_bounds__(256)
void gemm_wmma_f32(const float* __restrict__ A, int lda,
                   const float* __restrict__ W,
                   const float* __restrict__ bias,
                   float* __restrict__ C,
                   int M, int N, int K, int act)
{
    const int lane = threadIdx.x & 31;
    const int wave = threadIdx.x >> 5;
    const int ntM  = M >> 5;                 // 32-row macro tiles
    const int ntN  = (N + 63) >> 6;          // 64-col macro tiles
    const int tile = blockIdx.x * 8 + wave;
    if (tile >= ntM * ntN) return;

    const int tm = tile % ntM;               // consecutive waves share the W tile
    const int tn = tile / ntM;
    const long m0 = (long)tm * 32;
    const long n0 = (long)tn * 64;

    const int half = lane >> 4;              // 0: K=0,1 / rows M=0..7 ; 1: K=2,3 / M=8..15
    const int l16  = lane & 15;

    // A: lane reads rows (m0+l16) and (m0+16+l16), K pair selected by half.
    const float* ap0 = A + (m0 + l16) * (long)lda + 2 * half;
    const float* ap1 = ap0 + 16 * (long)lda;

    // W: 4 N-subtiles of 16; clamp row index so loads stay in bounds for tails.
    const float* wp[4];
#pragma unroll
    for (int j = 0; j < 4; ++j) {
        long n = n0 + 16 * j + l16;
        if (n > (long)N - 1) n = (long)N - 1;
        wp[j] = W + n * (long)K + 2 * half;
    }

    v8f acc0[4] = {v8f{}, v8f{}, v8f{}, v8f{}};
    v8f acc1[4] = {v8f{}, v8f{}, v8f{}, v8f{}};

#pragma unroll 4
    for (int k = 0; k < K; k += 4) {
        v2f a0 = *(const v2f*)(ap0 + k);
        v2f a1 = *(const v2f*)(ap1 + k);
        v2f b0 = *(const v2f*)(wp[0] + k);
        v2f b1 = *(const v2f*)(wp[1] + k);
        v2f b2 = *(const v2f*)(wp[2] + k);
        v2f b3 = *(const v2f*)(wp[3] + k);
        acc0[0] = __builtin_amdgcn_wmma_f32_16x16x4_f32(false, a0, false, b0,
                                                        (short)0, acc0[0], false, false);
        acc1[0] = __builtin_amdgcn_wmma_f32_16x16x4_f32(false, a1, false, b0,
                                                        (short)0, acc1[0], false, false);
        acc0[1] = __builtin_amdgcn_wmma_f32_16x16x4_f32(false, a0, false, b1,
                                                        (short)0, acc0[1], false, false);
        acc1[1] = __builtin_amdgcn_wmma_f32_16x16x4_f32(false, a1, false, b1,
                                                        (short)0, acc1[1], false, false);
        acc0[2] = __builtin_amdgcn_wmma_f32_16x16x4_f32(false, a0, false, b2,
                                                        (short)0, acc0[2], false, false);
        acc1[2] = __builtin_amdgcn_wmma_f32_16x16x4_f32(false, a1, false, b2,
                                                        (short)0, acc1[2], false, false);
        acc0[3] = __builtin_amdgcn_wmma_f32_16x16x4_f32(false, a0, false, b3,
                                                        (short)0, acc0[3], false, false);
        acc1[3] = __builtin_amdgcn_wmma_f32_16x16x4_f32(false, a1, false, b3,
                                                        (short)0, acc1[3], false, false);
    }

    // C/D layout: lanes 0-15: VGPR r -> (M=r, N=lane); lanes 16-31: M=8+r.
#pragma unroll
    for (int j = 0; j < 4; ++j) {
        long n = n0 + 16 * j + l16;
        if (n < (long)N) {
            float bv = bias ? bias[n] : 0.0f;
#pragma unroll
            for (int r = 0; r < 8; ++r) {
                float v0 = acc0[j][r] + bv;
                float v1 = acc1[j][r] + bv;
                if (act == 1) {
                    v0 = (v0 > 20.0f) ? v0 : log1pf(expf(v0));
                    v1 = (v1 > 20.0f) ? v1 : log1pf(expf(v1));
                }
                C[(m0 + r + 8 * half) * (long)N + n]      = v0;
                C[(m0 + 16 + r + 8 * half) * (long)N + n] = v1;
            }
        }
    }
}

// ---------------------------------------------------------------------------
// Audio front-end
// ---------------------------------------------------------------------------
__global__ __launch_bounds__(256)
void conv1_relu_kernel(const float* __restrict__ x,   // [B,80,1024]
                       const float* __restrict__ w,   // [192,80,3]
                       const float* __restrict__ bias,
                       float* __restrict__ y)         // [B,192,1024]
{
    int idx = blockIdx.x * 256 + threadIdx.x;
    if (idx >= BATCH * 192 * L_AUDIO) return;
    int l = idx & (L_AUDIO - 1);
    int o = (idx >> 10) % 192;
    int b = idx / (192 * L_AUDIO);
    float acc = bias[o];
    for (int i = 0; i < AUDIO_DIM; ++i) {
        const float* xr = x + ((long)b * AUDIO_DIM + i) * L_AUDIO;
        const float* wr = w + ((long)o * AUDIO_DIM + i) * 3;
        float x0 = (l >= 1) ? xr[l - 1] : 0.0f;
        float x1 = xr[l];
        float x2 = (l < L_AUDIO - 1) ? xr[l + 1] : 0.0f;
        acc = fmaf(wr[0], x0, acc);
        acc = fmaf(wr[1], x1, acc);
        acc = fmaf(wr[2], x2, acc);
    }
    y[idx] = fmaxf(acc, 0.0f);
}

__global__ __launch_bounds__(256)
void conv2_to_x_kernel(const float* __restrict__ a1,  // [B,192,1024]
                       const float* __restrict__ w,   // [384,192,3]
                       const float* __restrict__ bias,
                       float* __restrict__ X)         // [B,2048,384] (audio half)
{
    int idx = blockIdx.x * 256 + threadIdx.x;
    if (idx >= BATCH * D_MODEL * L_AUDIO) return;
    int l = idx & (L_AUDIO - 1);
    int o = (idx >> 10) % D_MODEL;
    int b = idx / (D_MODEL * L_AUDIO);
    float acc = bias[o];
    for (int i = 0; i < 192; ++i) {
        const float* xr = a1 + ((long)b * 192 + i) * L_AUDIO;
        const float* wr = w + ((long)o * 192 + i) * 3;
        float x0 = (l >= 1) ? xr[l - 1] : 0.0f;
        float x1 = xr[l];
        float x2 = (l < L_AUDIO - 1) ? xr[l + 1] : 0.0f;
        acc = fmaf(wr[0], x0, acc);
        acc = fmaf(wr[1], x1, acc);
        acc = fmaf(wr[2], x2, acc);
    }
    X[((long)b * L_TOT + l) * D_MODEL + o] = acc;   // transposed into [B,L,D]
}

__global__ __launch_bounds__(256)
void embed_kernel(const int* __restrict__ tok,       // [B,1024]
                  const float* __restrict__ embed,   // [VOCAB,384]
                  float* __restrict__ X)             // [B,2048,384] (text half)
{
    int idx = blockIdx.x * 256 + threadIdx.x;
    if (idx >= BATCH * L_TEXT * D_MODEL) return;
    int c = idx % D_MODEL;
    int l = (idx / D_MODEL) % L_TEXT;
    int b = idx / (D_MODEL * L_TEXT);
    int t = tok[b * L_TEXT + l];
    X[((long)b * L_TOT + (L_AUDIO + l)) * D_MODEL + c] = embed[(long)t * D_MODEL + c];
}

// ---------------------------------------------------------------------------
// Depthwise causal conv (K=4, left-pad 3) + SiLU on the xi half of xz
// ---------------------------------------------------------------------------
__global__ __launch_bounds__(256)
void dwconv_silu_kernel(const float* __restrict__ xz,   // [4096,1536]
                        const float* __restrict__ cw,   // [768,1,4]
                        const float* __restrict__ cb,   // [768]
                        float* __restrict__ xi)         // [4096,768]
{
    int idx = blockIdx.x * 256 + threadIdx.x;
    if (idx >= ROWS * D_INNER) return;
    int d = idx % D_INNER;
    long row = idx / D_INNER;
    int l = (int)(row & (L_TOT - 1));
    float acc = cb[d];
#pragma unroll
    for (int k = 0; k < D_CONV; ++k) {
        int ll = l - (D_CONV - 1) + k;
        float v = (ll >= 0) ? xz[(row + (ll - l)) * (2 * D_INNER) + d] : 0.0f;
        acc = fmaf(cw[d * D_CONV + k], v, acc);
    }
    xi[idx] = acc / (1.0f + expf(-acc));   // SiLU
}

// ---------------------------------------------------------------------------
// Selective scan. Lane = (channel d, state n); wave = 2 channels x 16 states.
// Sequential over L=2048; xor-shuffle reduce over the 16 states; lane n==0
// fuses the D skip + SiLU(z) gate.
// ---------------------------------------------------------------------------
__global__ __launch_bounds__(256)
void mamba_scan_kernel(const float* __restrict__ xz,    // [4096,1536] (z half)
                       const float* __restrict__ xi,    // [4096,768]
                       const float* __restrict__ dt,    // [4096,768]
                       const float* __restrict__ xdbl,  // [4096,56] (B at 24, C at 40)
                       const float* __restrict__ A_log, // [768,16]
                       const float* __restrict__ Dp,    // [768]
                       float* __restrict__ yg)          // [4096,768]
{
    const int n    = threadIdx.x & 15;
    const int chan = threadIdx.x >> 4;                 // 0..15 in block
    const int b    = blockIdx.x / (D_INNER / 16);      // 48 blocks per batch
    const int d    = (blockIdx.x % (D_INNER / 16)) * 16 + chan;

    const float Ad = -expf(A_log[d * D_STATE + n]);
    const float Dd = Dp[d];
    float h = 0.0f;
    const long rowbase = (long)b * L_TOT;

    for (int l = 0; l < L_TOT; ++l) {
        long row  = rowbase + l;
        float dtv = dt[row * D_INNER + d];
        float xiv = xi[row * D_INNER + d];
        float Bv  = xdbl[row * 56 + DT_RANK + n];
        float Cv  = xdbl[row * 56 + DT_RANK + D_STATE + n];

        float dA = expf(dtv * Ad);
        h = fmaf(dA, h, dtv * Bv * xiv);

        float p = h * Cv;
        p += __shfl_xor(p, 1, 32);
        p += __shfl_xor(p, 2, 32);
        p += __shfl_xor(p, 4, 32);
        p += __shfl_xor(p, 8, 32);

        if (n == 0) {
            float zv  = xz[row * (2 * D_INNER) + D_INNER + d];
            float sil = zv / (1.0f + expf(-zv));
            yg[row * D_INNER + d] = (p + xiv * Dd) * sil;
        }
    }
}

// ---------------------------------------------------------------------------
// Host-side launch
// ---------------------------------------------------------------------------
static inline int gblocks(long tiles) { return (int)((tiles + 7) / 8); }

static void run_mamba_block(const float* Xin, float* Xout,
                            const float* in_w, const float* convw, const float* convb,
                            const float* xproj_w, const float* dt_w, const float* dt_b,
                            const float* A_log, const float* Dp, const float* out_w,
                            float* xz, float* xi, float* xdbl, float* dtb, float* yg,
                            hipStream_t stream)
{
    // in_proj: [4096,384] @ [1536,384]^T -> xz[4096,1536]
    {
        long tiles = (ROWS / 32) * ((2 * D_INNER + 63) / 64);
        hipLaunchKernelGGL(gemm_wmma_f32, dim3(gblocks(tiles)), dim3(256), 0, stream,
                           Xin, D_MODEL, in_w, (const float*)nullptr, xz,
                           ROWS, 2 * D_INNER, D_MODEL, 0);
    }
    // depthwise causal conv + SiLU
    hipLaunchKernelGGL(dwconv_silu_kernel,
                       dim3((ROWS * D_INNER + 255) / 256), dim3(256), 0, stream,
                       xz, convw, convb, xi);
    // x_proj: [4096,768] @ [56,768]^T -> xdbl[4096,56]
    {
        long tiles = (ROWS / 32) * ((DT_RANK + 2 * D_STATE + 63) / 64);
        hipLaunchKernelGGL(gemm_wmma_f32, dim3(gblocks(tiles)), dim3(256), 0, stream,
                           xi, D_INNER, xproj_w, (const float*)nullptr, xdbl,
                           ROWS, DT_RANK + 2 * D_STATE, D_INNER, 0);
    }
    // dt_proj + bias + softplus: xdbl[:, :24] (lda=56) @ [768,24]^T -> dt[4096,768]
    {
        long tiles = (ROWS / 32) * ((D_INNER + 63) / 64);
        hipLaunchKernelGGL(gemm_wmma_f32, dim3(gblocks(tiles)), dim3(256), 0, stream,
                           xdbl, 56, dt_w, dt_b, dtb,
                           ROWS, D_INNER, DT_RANK, 1);
    }
    // selective scan + D skip + gate
    hipLaunchKernelGGL(mamba_scan_kernel,
                       dim3(BATCH * (D_INNER / 16)), dim3(256), 0, stream,
                       xz, xi, dtb, xdbl, A_log, Dp, yg);
    // out_proj: [4096,768] @ [384,768]^T -> Xout[4096,384]
    {
        long tiles = (ROWS / 32) * ((D_MODEL + 63) / 64);
        hipLaunchKernelGGL(gemm_wmma_f32, dim3(gblocks(tiles)), dim3(256), 0, stream,
                           yg, D_INNER, out_w, (const float*)nullptr, Xout,
                           ROWS, D_MODEL, D_INNER, 0);
    }
}

extern "C" void kernel_launch(void* const* d_in, const int* in_sizes, int n_in,
                              void* d_out, int out_size, void* d_ws, size_t ws_size,
                              hipStream_t stream)
{
    const float* audio   = (const float*)d_in[0];
    const int*   tokens  = (const int*)d_in[1];
    const float* conv1_w = (const float*)d_in[2];
    const float* conv1_b = (const float*)d_in[3];
    const float* conv2_w = (const float*)d_in[4];
    const float* conv2_b = (const float*)d_in[5];
    const float* embed_w = (const float*)d_in[6];
    const float* m1[9]   = {(const float*)d_in[7],  (const float*)d_in[8],
                            (const float*)d_in[9],  (const float*)d_in[10],
                            (const float*)d_in[11], (const float*)d_in[12],
                            (const float*)d_in[13], (const float*)d_in[14],
                            (const float*)d_in[15]};
    const float* m2[9]   = {(const float*)d_in[16], (const float*)d_in[17],
                            (const float*)d_in[18], (const float*)d_in[19],
                            (const float*)d_in[20], (const float*)d_in[21],
                            (const float*)d_in[22], (const float*)d_in[23],
                            (const float*)d_in[24]};
    const float* proj_w  = (const float*)d_in[25];
    const float* proj_b  = (const float*)d_in[26];
    float* out = (float*)d_out;

    // Workspace carve-up (floats): ~78 MB total
    float* ws = (float*)d_ws;
    size_t o = 0;
    float* a1   = ws + o; o += (size_t)BATCH * 192 * L_AUDIO;        // conv1 out
    float* X    = ws + o; o += (size_t)ROWS * D_MODEL;               // activations A
    float* X2   = ws + o; o += (size_t)ROWS * D_MODEL;               // activations B
    float* xz   = ws + o; o += (size_t)ROWS * 2 * D_INNER;
    float* xi   = ws + o; o += (size_t)ROWS * D_INNER;
    float* xdbl = ws + o; o += (size_t)ROWS * 56;
    float* dtb  = ws + o; o += (size_t)ROWS * D_INNER;
    float* yg   = ws + o; o += (size_t)ROWS * D_INNER;
    (void)ws_size; (void)n_in; (void)in_sizes; (void)out_size;

    // Audio front-end + embedding -> X [B, 2048, 384]
    hipLaunchKernelGGL(conv1_relu_kernel,
                       dim3((BATCH * 192 * L_AUDIO + 255) / 256), dim3(256), 0, stream,
                       audio, conv1_w, conv1_b, a1);
    hipLaunchKernelGGL(conv2_to_x_kernel,
                       dim3((BATCH * D_MODEL * L_AUDIO + 255) / 256), dim3(256), 0, stream,
                       a1, conv2_w, conv2_b, X);
    hipLaunchKernelGGL(embed_kernel,
                       dim3((BATCH * L_TEXT * D_MODEL + 255) / 256), dim3(256), 0, stream,
                       tokens, embed_w, X);

    // Two Mamba blocks (no residual in reference)
    run_mamba_block(X,  X2, m1[0], m1[1], m1[2], m1[3], m1[4], m1[5], m1[6], m1[7], m1[8],
                    xz, xi, xdbl, dtb, yg, stream);
    run_mamba_block(X2, X,  m2[0], m2[1], m2[2], m2[3], m2[4], m2[5], m2[6], m2[7], m2[8],
                    xz, xi, xdbl, dtb, yg, stream);

    // Final vocab projection: [4096,384] @ [50257,384]^T + bias -> out
    {
        long tiles = (long)(ROWS / 32) * ((VOCABSZ + 63) / 64);
        hipLaunchKernelGGL(gemm_wmma_f32, dim3(gblocks(tiles)), dim3(256), 0, stream,
                           X, D_MODEL, proj_w, proj_b, out,
                           ROWS, VOCABSZ, D_MODEL, 0);
    }
}